// DynamicMaskHead_74294344286901
// MI455X (gfx1250) — compile-verified
//
#include <hip/hip_runtime.h>

// CDNA5 / gfx1250 — wave32, WMMA f32 16x16x4 path, padded-LDS operands,
// 10 tiles (160 px) per wave with hoisted A-operands, branchless loop body.
typedef __attribute__((ext_vector_type(2))) float v2f;
typedef __attribute__((ext_vector_type(8))) float v8f;

#define IN_CH   8
#define CH      8
#define HDIM    136
#define WDIM    200
#define HW      (HDIM * WDIM)          // 27200
#define N_INST  128
#define NUM_PARAMS 169
#define TILES_PER_INST (HW / 16)       // 1700
#define T_PER_WAVE 10                  // 1700 = 10 * 170 (exact)
#define WAVES_PER_BLOCK 8
#define WAVES_PER_INST (TILES_PER_INST / T_PER_WAVE)                    // 170
#define BLOCKS_X ((WAVES_PER_INST + WAVES_PER_BLOCK - 1) / WAVES_PER_BLOCK)  // 22

__global__ __launch_bounds__(256) void DynamicMaskHead_wmma_kernel(
    const float* __restrict__ mask_feats,      // (2, 8, 136, 200)
    const float* __restrict__ params,          // (128, 169)
    const float* __restrict__ inst_loc,        // (128, 2)
    const float* __restrict__ soi_table,       // (5,)
    const int*   __restrict__ im_inds,         // (128,)
    const int*   __restrict__ fpn_levels,      // (128,)
    const int*   __restrict__ stride_p,        // scalar
    float*       __restrict__ out)             // (128, 1, 136, 200)
{
    // Pre-padded parameter staging: all out-of-range entries are real zeros,
    // so every per-lane LDS read below is unconditional (no exec divergence).
    __shared__ float sW0[16 * 12];   // W0 padded 16x12 (valid 8x10)
    __shared__ float sW1[16 * 8];    // W1 padded 16x8  (valid 8x8)
    __shared__ float sB0[16];        // b0 padded (valid 8)
    __shared__ float sB1[16];        // b1 padded (valid 8)
    __shared__ float sW2[8];
    __shared__ float sB2;

    const int inst = blockIdx.y;
    const int tid  = threadIdx.x;
    const float* pp = params + (size_t)inst * NUM_PARAMS;

    for (int i = tid; i < 16 * 12; i += 256) {
        const int m = i / 12, k = i - m * 12;
        sW0[i] = (m < CH && k < 10) ? pp[m * 10 + k] : 0.0f;
    }
    if (tid < 128) {
        const int m = tid >> 3, k = tid & 7;
        sW1[tid] = (m < CH) ? pp[80 + m * 8 + k] : 0.0f;
    } else if (tid < 144) {
        const int r = tid - 128;
        sB0[r] = (r < CH) ? pp[152 + r] : 0.0f;
    } else if (tid < 160) {
        const int r = tid - 144;
        sB1[r] = (r < CH) ? pp[160 + r] : 0.0f;
    } else if (tid < 168) {
        sW2[tid - 160] = pp[144 + (tid - 160)];
    } else if (tid == 168) {
        sB2 = pp[168];
    }
    __syncthreads();

    const int lane  = tid & 31;
    const int wave  = tid >> 5;
    const int gwave = blockIdx.x * WAVES_PER_BLOCK + wave;
    if (gwave >= WAVES_PER_INST) return;   // wave-uniform; EXEC all-1s below

    const int n  = lane & 15;   // N column (pixel in tile) == A's M row
    const int hi = lane >> 4;   // lane-half selector
    const bool ishi = (hi != 0);

    // ---- hoisted per-instance scalars --------------------------------------
    const int   stride  = stride_p[0];
    const float inv_soi = 1.0f / soi_table[fpn_levels[inst]];
    const float locx    = inst_loc[inst * 2 + 0];
    const float locy    = inst_loc[inst * 2 + 1];
    const float* feats  = mask_feats + (size_t)im_inds[inst] * IN_CH * HW;

    // ---- hoisted A-operands (WMMA A layout: lane = M, K = c*4 + v + 2*hi) --
    v2f a0[3], a1[2];
#pragma unroll
    for (int c = 0; c < 3; ++c)
        a0[c] = *(const v2f*)(sW0 + n * 12 + c * 4 + 2 * hi);  // 8B-aligned
#pragma unroll
    for (int c = 0; c < 2; ++c)
        a1[c] = *(const v2f*)(sW1 + n * 8 + c * 4 + 2 * hi);

    // ---- hoisted biases (padded => no conditionals) and layer-2 weights ----
    float bb0[8], bb1[8], w2r[8];
#pragma unroll
    for (int r = 0; r < 8; ++r) { bb0[r] = sB0[8 * hi + r]; bb1[r] = sB1[8 * hi + r]; }
#pragma unroll
    for (int k = 0; k < 8; ++k) w2r[k] = sW2[k];
    const float b2 = sB2;

    // ---- hoisted B-operand feat-row element offsets: row = (K-2) & 7 -------
    // Every offset is in-bounds, so all loads below can be unconditional.
    int foff[3][2];
#pragma unroll
    for (int c = 0; c < 3; ++c)
#pragma unroll
        for (int v = 0; v < 2; ++v)
            foff[c][v] = (((c * 4 + v + 2 * hi) - 2) & 7) * HW;

    // ---- incremental pixel/coordinate state --------------------------------
    int p  = gwave * (T_PER_WAVE * 16) + n;
    int px = p % WDIM;
    float cx = (float)(px * stride + (stride >> 1));
    float cy = (float)((p / WDIM) * stride + (stride >> 1));
    const float step_cx  = (float)(16 * stride);
    const float wrap_dcx = (float)((16 - WDIM) * stride);
    const float step_cy  = (float)stride;

#pragma unroll 2
    for (int t = 0; t < T_PER_WAVE; ++t) {
        const float relx = (locx - cx) * inv_soi;
        const float rely = (locy - cy) * inv_soi;

        // Unconditional coalesced loads (no exec divergence); validity is
        // resolved afterwards with cndmask selects.
        const float fv00 = feats[p + foff[0][0]];
        const float fv01 = feats[p + foff[0][1]];
        const float fv10 = feats[p + foff[1][0]];
        const float fv11 = feats[p + foff[1][1]];
        const float fv20 = feats[p + foff[2][0]];
        const float fv21 = feats[p + foff[2][1]];

        // B0 (12pad x 16): chunk0 = coords(lo)/feats01(hi); chunk1 = feats2-5;
        // chunk2 = feats67(lo)/zero-pad(hi).
        v2f b0[3];
        b0[0][0] = ishi ? fv00 : relx;
        b0[0][1] = ishi ? fv01 : rely;
        b0[1][0] = fv10;
        b0[1][1] = fv11;
        b0[2][0] = ishi ? 0.0f : fv20;
        b0[2][1] = ishi ? 0.0f : fv21;

        v8f acc = {};
#pragma unroll
        for (int c = 0; c < 3; ++c)
            acc = __builtin_amdgcn_wmma_f32_16x16x4_f32(
                false, a0[c], false, b0[c], (short)0, acc, false, false);

        // bias + relu (padded biases: rows >= 8 stay exactly zero)
        float x1[8];
#pragma unroll
        for (int r = 0; r < 8; ++r)
            x1[r] = fmaxf(acc[r] + bb0[r], 0.0f);

        // Layer 1: D-layout -> B-layout. lanes 0-15: same lane, row K;
        // lanes 16-31: row K+2 fetched from lane (lane-16) via bpermute.
        v2f b1[2];
#pragma unroll
        for (int c = 0; c < 2; ++c)
#pragma unroll
            for (int v = 0; v < 2; ++v) {
                const float t_hi = __shfl(x1[c * 4 + v + 2], n, 32);
                b1[c][v] = ishi ? t_hi : x1[c * 4 + v];
            }

        v8f acc1 = {};
#pragma unroll
        for (int c = 0; c < 2; ++c)
            acc1 = __builtin_amdgcn_wmma_f32_16x16x4_f32(
                false, a1[c], false, b1[c], (short)0, acc1, false, false);

        float x2[8];
#pragma unroll
        for (int r = 0; r < 8; ++r)
            x2[r] = fmaxf(acc1[r] + bb1[r], 0.0f);

        // Layer 2: 1x8 dot, valid in lanes 0-15
        float o = b2;
#pragma unroll
        for (int k = 0; k < 8; ++k)
            o = fmaf(w2r[k], x2[k], o);

        if (!ishi)
            out[(size_t)inst * HW + p] = o;   // coalesced 64B store

        // advance 16 pixels without integer division
        p += 16;
        px += 16;
        const bool wrap = (px >= WDIM);
        px = wrap ? px - WDIM : px;
        cx = wrap ? cx + wrap_dcx : cx + step_cx;
        cy = wrap ? cy + step_cy : cy;
    }
}

extern "C" void kernel_launch(void* const* d_in, const int* in_sizes, int n_in,
                              void* d_out, int out_size, void* d_ws, size_t ws_size,
                              hipStream_t stream) {
    const float* mask_feats = (const float*)d_in[0];
    const float* params     = (const float*)d_in[1];
    const float* inst_loc   = (const float*)d_in[2];
    const float* soi_table  = (const float*)d_in[3];
    const int*   im_inds    = (const int*)d_in[4];
    const int*   fpn_levels = (const int*)d_in[5];
    const int*   stride_p   = (const int*)d_in[6];
    float*       out        = (float*)d_out;

    dim3 block(256, 1, 1);
    dim3 grid(BLOCKS_X, N_INST, 1);
    DynamicMaskHead_wmma_kernel<<<grid, block, 0, stream>>>(
        mask_feats, params, inst_loc, soi_table, im_inds, fpn_levels, stride_p, out);
}